// HammingLoss_90125593739862
// MI455X (gfx1250) — compile-verified
//
#include <hip/hip_runtime.h>
#include <stdint.h>

#define TPB 256
#define MAX_CLASSES 1024   // LDS table capacity (8 KB); actual n_classes = 1000
#define CODE_LEN 64

// b64 async-copy operand type per hipcc diagnostic: vector of 2 ints
typedef int v2i __attribute__((vector_size(8)));
typedef __attribute__((address_space(1))) v2i g_v2i;   // global
typedef __attribute__((address_space(3))) v2i l_v2i;   // LDS

// ---------------------------------------------------------------------------
// Kernel 1: pack coding_table rows (64 x fp32 in {0,1}) into uint64 bitmasks.
// Also zero the global accumulator (must happen every call).
// ---------------------------------------------------------------------------
__global__ void __launch_bounds__(TPB)
pack_kernel(const float* __restrict__ table,
            uint64_t* __restrict__ packed,
            unsigned* __restrict__ acc,
            int n_classes) {
    int c = blockIdx.x * TPB + threadIdx.x;
    if (c == 0) *acc = 0u;
    if (c >= n_classes) return;
    const float* row = table + (size_t)c * CODE_LEN;
    uint32_t lo = 0u, hi = 0u;
#pragma unroll
    for (int i = 0; i < 32; ++i) lo |= (row[i]      != 0.0f ? 1u : 0u) << i;
#pragma unroll
    for (int i = 0; i < 32; ++i) hi |= (row[32 + i] != 0.0f ? 1u : 0u) << i;
    packed[c] = ((uint64_t)hi << 32) | (uint64_t)lo;
}

// ---------------------------------------------------------------------------
// Kernel 2: stream index pairs, xor/popcount against LDS-resident table.
// Table staged into LDS with CDNA5 async global->LDS DMA when available.
// ---------------------------------------------------------------------------
__global__ void __launch_bounds__(TPB)
hamming_kernel(const int* __restrict__ outv,
               const int* __restrict__ tgtv,
               const uint64_t* __restrict__ packed,
               unsigned* __restrict__ acc,
               int n_classes, int n4, int B) {
    __shared__ uint64_t lds_codes[MAX_CLASSES];
    __shared__ unsigned red[TPB];

    const int t = threadIdx.x;

    // ---- Stage packed table (8 KB) into LDS ----
#if __has_builtin(__builtin_amdgcn_global_load_async_to_lds_b64)
    for (int k = t; k < n_classes; k += TPB) {
        __builtin_amdgcn_global_load_async_to_lds_b64(
            (g_v2i*)(packed + k),
            (l_v2i*)(&lds_codes[k]),
            /*offset=*/0, /*cpol=*/0);
    }
#if __has_builtin(__builtin_amdgcn_s_wait_asynccnt)
    __builtin_amdgcn_s_wait_asynccnt(0);
#else
    asm volatile("s_wait_asynccnt 0" ::: "memory");
#endif
#else
    for (int k = t; k < n_classes; k += TPB) lds_codes[k] = packed[k];
#endif
    __syncthreads();

    // ---- Stream indices, accumulate popcounts ----
    unsigned local = 0u;
    const int4* o4 = (const int4*)outv;
    const int4* g4 = (const int4*)tgtv;
    const int stride = gridDim.x * TPB;
    int gid = blockIdx.x * TPB + t;

    for (int i = gid; i < n4; i += stride) {
        int4 a = o4[i];
        int4 b = g4[i];
        local += (unsigned)__builtin_popcountll(lds_codes[a.x] ^ lds_codes[b.x]);
        local += (unsigned)__builtin_popcountll(lds_codes[a.y] ^ lds_codes[b.y]);
        local += (unsigned)__builtin_popcountll(lds_codes[a.z] ^ lds_codes[b.z]);
        local += (unsigned)__builtin_popcountll(lds_codes[a.w] ^ lds_codes[b.w]);
    }
    // Tail (B not divisible by 4) — empty for B = 2^21 but kept for safety.
    for (int i = n4 * 4 + gid; i < B; i += stride) {
        local += (unsigned)__builtin_popcountll(lds_codes[outv[i]] ^ lds_codes[tgtv[i]]);
    }

    // ---- Block reduction + one atomic per block ----
    red[t] = local;
    __syncthreads();
#pragma unroll
    for (int s = TPB / 2; s > 0; s >>= 1) {
        if (t < s) red[t] += red[t + s];
        __syncthreads();
    }
    if (t == 0) atomicAdd(acc, red[0]);
}

// ---------------------------------------------------------------------------
// Kernel 3: finalize scalar loss = acc / B
// ---------------------------------------------------------------------------
__global__ void finalize_kernel(const unsigned* __restrict__ acc,
                                float* __restrict__ out, float invB) {
    out[0] = (float)(*acc) * invB;
}

// ---------------------------------------------------------------------------
extern "C" void kernel_launch(void* const* d_in, const int* in_sizes, int n_in,
                              void* d_out, int out_size, void* d_ws, size_t ws_size,
                              hipStream_t stream) {
    const int*   outv  = (const int*)d_in[0];
    const int*   tgtv  = (const int*)d_in[1];
    const float* table = (const float*)d_in[2];

    const int B = in_sizes[0];
    int n_classes = in_sizes[2] / CODE_LEN;       // 1000
    if (n_classes > MAX_CLASSES) n_classes = MAX_CLASSES;

    // Workspace layout: [0, 8 KB) packed uint64 codes; [8 KB] u32 accumulator.
    uint64_t* packed = (uint64_t*)d_ws;
    unsigned* acc    = (unsigned*)((char*)d_ws + MAX_CLASSES * sizeof(uint64_t));

    // 1) pack table + zero accumulator
    int pack_blocks = (n_classes + TPB - 1) / TPB;
    pack_kernel<<<pack_blocks, TPB, 0, stream>>>(table, packed, acc, n_classes);

    // 2) main streaming pass (int4 groups)
    const int n4 = B / 4;                          // 524288 for B = 2^21
    int blocks = (n4 + TPB - 1) / TPB;
    if (blocks > 1024) blocks = 1024;              // grid-stride: 2 iters/thread
    if (blocks < 1) blocks = 1;
    hamming_kernel<<<blocks, TPB, 0, stream>>>(outv, tgtv, packed, acc,
                                               n_classes, n4, B);

    // 3) finalize
    finalize_kernel<<<1, 1, 0, stream>>>(acc, (float*)d_out, 1.0f / (float)B);
}